// Attention_62603443306616
// MI455X (gfx1250) — compile-verified
//
#include <hip/hip_runtime.h>
#include <math.h>

// ---------------- problem constants ----------------
#define B_DIM   64
#define S_DIM   4096
#define H_DIM   1024
#define NSPLIT  16
#define ROWS_PER_BLOCK (S_DIM / NSPLIT)   // 256 rows per block
#define CHUNK   32                        // rows per LDS chunk
#define NCHUNK  (ROWS_PER_BLOCK / CHUNK)  // 8
#define THREADS 256                       // 8 wave32s
#define NWAVES  (THREADS / 32)
#define STRIDE  1028                      // padded LDS row stride (floats) -> conflict-free
#define PART_STRIDE (H_DIM + 4)           // [m, l, pad, pad, c[1024]] (keeps c 16B aligned)

typedef float v2f __attribute__((ext_vector_type(2)));
typedef float v8f __attribute__((ext_vector_type(8)));
typedef int   v4i __attribute__((vector_size(16)));   // matches builtin param type

#define AS1 __attribute__((address_space(1)))
#define AS3 __attribute__((address_space(3)))

// ---------------- CDNA5 async global->LDS copy (guarded) ----------------
__device__ __forceinline__ void async_copy_f4(const float* g, float* l) {
#if __has_builtin(__builtin_amdgcn_global_load_async_to_lds_b128)
  float* gp = const_cast<float*>(g);
  __builtin_amdgcn_global_load_async_to_lds_b128(
      (AS1 v4i*)gp, (AS3 v4i*)l, 0, 0);
#else
  *(float4*)l = *(const float4*)g;   // sync fallback: global_load_b128 + ds_store_b128
#endif
}

__device__ __forceinline__ void wait_async_done() {
#if __has_builtin(__builtin_amdgcn_global_load_async_to_lds_b128)
#if __has_builtin(__builtin_amdgcn_s_wait_asynccnt)
  __builtin_amdgcn_s_wait_asynccnt(0);
#else
  asm volatile("s_wait_asynccnt 0" ::: "memory");
#endif
#endif
}

// ---------------- kernel 1: fused scores + online softmax + weighted sum ----
__global__ __launch_bounds__(THREADS)
void attn_partial_kernel(const float* __restrict__ hidden,
                         const float* __restrict__ enc,
                         float* __restrict__ part) {
  extern __shared__ float smem[];
  float* buf0    = smem;                       // CHUNK*STRIDE floats (128.5 KB)
  float* buf1    = smem + CHUNK * STRIDE;      // double buffer
  float* hs      = smem + 2 * CHUNK * STRIDE;  // hidden[b], 1024 floats
  float* scratch = hs + H_DIM;                 // NWAVES*16 wave score partials
  float* sarr    = scratch + NWAVES * 16;      // CHUNK chunk scores
  float* warr    = sarr + CHUNK;               // CHUNK exp-weights

  const int t  = threadIdx.x;
  const int b  = blockIdx.x / NSPLIT;
  const int sp = blockIdx.x % NSPLIT;
  const int s0 = sp * ROWS_PER_BLOCK;
  const float* encB = enc + (size_t)b * S_DIM * H_DIM;

  // stage hidden[b] into LDS (1024 floats, 4 per thread)
  *(float4*)(hs + 4 * t) = *(const float4*)(hidden + (size_t)b * H_DIM + 4 * t);

  // issue async loads for chunk 0: thread t loads cols [4t,4t+3] of every row
  for (int j = 0; j < CHUNK; ++j)
    async_copy_f4(encB + (size_t)(s0 + j) * H_DIM + 4 * t, buf0 + j * STRIDE + 4 * t);

  const int wave  = t >> 5;
  const int lane  = t & 31;
  const int g16   = wave & 1;        // 16-row group (0 or 1)
  const int slice = wave >> 1;       // K slice (0..3), 256 wide
  const int arow  = (g16 * 16 + (lane & 15)) * STRIDE;  // A-matrix row for this lane
  const int kk    = 2 * (lane >> 4);                    // K sub-offset per WMMA layout

  float m = -3.0e38f, lsum = 0.f;
  float c0 = 0.f, c1 = 0.f, c2 = 0.f, c3 = 0.f;

  for (int ch = 0; ch < NCHUNK; ++ch) {
    float* cur = (ch & 1) ? buf1 : buf0;
    wait_async_done();
    __syncthreads();                                   // (A) chunk ready, prev compute done

    if (ch + 1 < NCHUNK) {                             // prefetch next chunk
      float* nxt = (ch & 1) ? buf0 : buf1;
      const int sb = s0 + (ch + 1) * CHUNK;
      for (int j = 0; j < CHUNK; ++j)
        async_copy_f4(encB + (size_t)(sb + j) * H_DIM + 4 * t, nxt + j * STRIDE + 4 * t);
    }

    // ---- scores: per-wave 16-row x 256-K tile via V_WMMA_F32_16X16X4_F32 ----
#if __has_builtin(__builtin_amdgcn_wmma_f32_16x16x4_f32)
    {
      v8f acc = {};
      const float* ap = cur + arow + kk;   // A: E rows (16x4 per step)
      const float* bp = hs + kk;           // B: hidden replicated across N columns
      const int kbase = slice * 256;
      #pragma unroll 8
      for (int k0 = kbase; k0 < kbase + 256; k0 += 4) {
        v2f a  = *(const v2f*)(ap + k0);
        v2f bb = *(const v2f*)(bp + k0);
        acc = __builtin_amdgcn_wmma_f32_16x16x4_f32(
            false, a, false, bb, (short)0, acc, false, false);
      }
      // D layout: lane l, VGPR j -> row M = j + 8*(l/16); columns replicated
      if ((lane & 15) == 0) {
        const int half = lane >> 4;
        #pragma unroll
        for (int j = 0; j < 8; ++j)
          scratch[wave * 16 + half * 8 + j] = acc[j];
      }
    }
#else
    if ((lane & 16) == 0) {                // scalar fallback (never expected)
      const float* rp = cur + (g16 * 16 + (lane & 15)) * STRIDE;
      const int kbase = slice * 256;
      float s = 0.f;
      for (int k0 = kbase; k0 < kbase + 256; ++k0) s = fmaf(rp[k0], hs[k0], s);
      scratch[wave * 16 + (lane & 15)] = s;
    }
#endif
    __syncthreads();                                   // (B)

    if (t < CHUNK) {                                   // reduce 4 K-slice partials
      const int gg = t >> 4, idx = t & 15;
      float s = 0.f;
      #pragma unroll
      for (int sl = 0; sl < 4; ++sl) s += scratch[(2 * sl + gg) * 16 + idx];
      sarr[t] = s * (1.0f / 32.0f);                    // 1/sqrt(1024)
    }
    __syncthreads();                                   // (C)

    // online softmax update (uniform, computed redundantly per thread)
    float mc = -3.0e38f;
    #pragma unroll
    for (int s2 = 0; s2 < CHUNK; ++s2) mc = fmaxf(mc, sarr[s2]);
    const float mn = fmaxf(m, mc);
    const float sc = __expf(m - mn);
    if (t < CHUNK) warr[t] = __expf(sarr[t] - mn);
    __syncthreads();                                   // (D)

    float add = 0.f;
    #pragma unroll
    for (int s2 = 0; s2 < CHUNK; ++s2) add += warr[s2];
    lsum = lsum * sc + add;
    m = mn;
    c0 *= sc; c1 *= sc; c2 *= sc; c3 *= sc;

    // weighted accumulation from LDS (thread t owns h = 4t..4t+3)
    const float* ep = cur + 4 * t;
    #pragma unroll
    for (int s2 = 0; s2 < CHUNK; ++s2) {
      const float4 e = *(const float4*)(ep + s2 * STRIDE);
      const float wv = warr[s2];
      c0 = fmaf(wv, e.x, c0); c1 = fmaf(wv, e.y, c1);
      c2 = fmaf(wv, e.z, c2); c3 = fmaf(wv, e.w, c3);
    }
  }

  float* po = part + (size_t)blockIdx.x * PART_STRIDE;
  if (t == 0) { po[0] = m; po[1] = lsum; }
  *(float4*)(po + 4 + 4 * t) = make_float4(c0, c1, c2, c3);
}

// ---------------- kernel 2: merge NSPLIT partials per batch ----------------
__global__ __launch_bounds__(THREADS)
void attn_combine_kernel(const float* __restrict__ part, float* __restrict__ out) {
  const int b = blockIdx.x;
  const int t = threadIdx.x;
  const float* pb = part + (size_t)b * NSPLIT * PART_STRIDE;

  float M = -3.0e38f;
  #pragma unroll
  for (int i = 0; i < NSPLIT; ++i) M = fmaxf(M, pb[(size_t)i * PART_STRIDE]);
  float f[NSPLIT];
  float L = 0.f;
  #pragma unroll
  for (int i = 0; i < NSPLIT; ++i) {
    f[i] = __expf(pb[(size_t)i * PART_STRIDE] - M);
    L = fmaf(pb[(size_t)i * PART_STRIDE + 1], f[i], L);
  }
  const float inv = 1.0f / L;

  float a0 = 0.f, a1 = 0.f, a2 = 0.f, a3 = 0.f;
  #pragma unroll
  for (int i = 0; i < NSPLIT; ++i) {
    const float4 ci = *(const float4*)(pb + (size_t)i * PART_STRIDE + 4 + 4 * t);
    a0 = fmaf(f[i], ci.x, a0); a1 = fmaf(f[i], ci.y, a1);
    a2 = fmaf(f[i], ci.z, a2); a3 = fmaf(f[i], ci.w, a3);
  }
  *(float4*)(out + (size_t)b * H_DIM + 4 * t) =
      make_float4(a0 * inv, a1 * inv, a2 * inv, a3 * inv);
}

// ---------------- launch ----------------
extern "C" void kernel_launch(void* const* d_in, const int* in_sizes, int n_in,
                              void* d_out, int out_size, void* d_ws, size_t ws_size,
                              hipStream_t stream) {
  (void)in_sizes; (void)n_in; (void)out_size; (void)ws_size;
  const float* hidden = (const float*)d_in[0];   // [64, 1024] f32
  const float* enc    = (const float*)d_in[1];   // [64, 4096, 1024] f32
  float* out  = (float*)d_out;                   // [64, 1024] f32
  float* part = (float*)d_ws;                    // 1024 * (1024+4) floats ~ 4.2 MB

  const size_t lds_bytes =
      (size_t)(2 * CHUNK * STRIDE + H_DIM + NWAVES * 16 + 2 * CHUNK) * sizeof(float); // ~268 KB

  (void)hipFuncSetAttribute((const void*)attn_partial_kernel,
                            hipFuncAttributeMaxDynamicSharedMemorySize, (int)lds_bytes);

  attn_partial_kernel<<<dim3(B_DIM * NSPLIT), dim3(THREADS), lds_bytes, stream>>>(
      hidden, enc, part);
  attn_combine_kernel<<<dim3(B_DIM), dim3(THREADS), 0, stream>>>(part, out);
}